// GHMC_Loss_28956669509642
// MI455X (gfx1250) — compile-verified
//
#include <hip/hip_runtime.h>
#include <stdint.h>

#define BINS 10
#define BLK  256
#define C    1000          // classes per row (4000 bytes)
#define NB1  4096          // pass1 grid
#define NB2  256           // pass2 grid

// ---------------------------------------------------------------------------
// Async global -> LDS copy of one float4 per active lane (GVS addressing:
// saddr = 64-bit base in SGPR pair, vaddr = 32-bit byte offset, vdst = LDS
// byte address). Tracked with ASYNCcnt (in-order completion per wave).
// ---------------------------------------------------------------------------
__device__ __forceinline__ void async_ld128(uint32_t ldsAddr, uint32_t gOff,
                                            const float* base) {
    asm volatile("global_load_async_to_lds_b128 %0, %1, %2 offset:0"
                 :: "v"(ldsAddr), "v"(gOff), "s"(base)
                 : "memory");
}
__device__ __forceinline__ void wait_async_le4() {
    asm volatile("s_wait_asynccnt 0x4" ::: "memory");
}
__device__ __forceinline__ void wait_async_le2() {
    asm volatile("s_wait_asynccnt 0x2" ::: "memory");
}
__device__ __forceinline__ void wait_async_le0() {
    asm volatile("s_wait_asynccnt 0x0" ::: "memory");
}

// ---------------------------------------------------------------------------
// Kernel 1 (first in file so the disasm snippet shows it): per-row softmax
// stats via triple-buffered async LDS staging (2 rows of prefetch depth).
// Emits ce[r], bin[r]; accumulates histogram with integer atomics.
// ---------------------------------------------------------------------------
__global__ __launch_bounds__(BLK) void ghmc_pass1(
    const float* __restrict__ input, const float* __restrict__ target,
    float* __restrict__ ce_out, int* __restrict__ bin_out,
    unsigned* __restrict__ gcounts, int B)
{
    __shared__ __align__(16) float sIn[3][1024];   // 12 KB
    __shared__ __align__(16) float sTg[3][1024];   // 12 KB
    __shared__ float redA[8];
    __shared__ float redB[8];
    __shared__ unsigned shist[BINS];

    const int t = threadIdx.x;
    if (t < BINS) shist[t] = 0u;

    const uint32_t ldsIn = (uint32_t)(uintptr_t)(&sIn[0][0]);
    const uint32_t ldsTg = (uint32_t)(uintptr_t)(&sTg[0][0]);

    const bool act = (t < 250);                 // 250 float4 per 1000-float row
    const uint32_t tb = (uint32_t)t * 16u;

    auto issue_row = [&](int r, int buf) {
        if (act) {
            uint32_t go = (uint32_t)r * (uint32_t)(C * 4) + tb;
            uint32_t bo = (uint32_t)buf * 4096u + tb;
            async_ld128(ldsIn + bo, go, input);
            async_ld128(ldsTg + bo, go, target);
        }
    };

    const int stride = gridDim.x;
    int r = blockIdx.x;
    if (r < B)          issue_row(r, 0);            // prefetch depth 2
    if (r + stride < B) issue_row(r + stride, 1);
    int cur = 0;

    for (; r < B; r += stride) {
        const long rn2 = (long)r + 2L * stride;
        if (rn2 < (long)B) {
            int b2 = cur + 2; if (b2 >= 3) b2 -= 3;
            issue_row((int)rn2, b2);
            wait_async_le4();                   // row r's 2 copies complete
        } else if (r + stride < B) {
            wait_async_le2();
        } else {
            wait_async_le0();
        }
        __syncthreads();                        // all waves' copies visible

        const float* bin = sIn[cur];
        const float* btg = sTg[cur];

        // Pass A: row max and x[label] = dot(one_hot, x); stage x in VGPRs.
        float xv[4];
        float m = -3.402823466e38f, xl = 0.0f;
        #pragma unroll
        for (int k = 0; k < 4; k++) {
            int i = t + k * BLK;
            if (i < C) {
                float x = bin[i];
                xv[k] = x;
                m = fmaxf(m, x);
                xl = fmaf(btg[i], x, xl);
            } else {
                xv[k] = -3.402823466e38f;       // exp(-huge) == 0
            }
        }
        #pragma unroll
        for (int o = 16; o > 0; o >>= 1) {
            m  = fmaxf(m, __shfl_xor(m, o, 32));
            xl += __shfl_xor(xl, o, 32);
        }
        if ((t & 31) == 0) { redA[t >> 5] = m; redB[t >> 5] = xl; }
        __syncthreads();
        m = redA[0]; xl = redB[0];
        #pragma unroll
        for (int i = 1; i < 8; i++) { m = fmaxf(m, redA[i]); xl += redB[i]; }
        __syncthreads();

        // Pass B: Z = sum exp(x - m) from registers (no LDS re-read)
        float z = 0.0f;
        #pragma unroll
        for (int k = 0; k < 4; k++) {
            int i = t + k * BLK;
            if (i < C) z += expf(xv[k] - m);
        }
        #pragma unroll
        for (int o = 16; o > 0; o >>= 1) z += __shfl_xor(z, o, 32);
        if ((t & 31) == 0) redA[t >> 5] = z;
        __syncthreads();
        z = redA[0];
        #pragma unroll
        for (int i = 1; i < 8; i++) z += redA[i];

        if (t == 0) {
            float ce = logf(z) + m - xl;                    // -log_softmax[label]
            float g  = fabsf(1.0f - expf(xl - m) / z);      // |p_label - 1|
            int bi = (int)(g * 10.0f);                      // trunc == astype(i32)
            bi = bi < 0 ? 0 : (bi > BINS - 1 ? BINS - 1 : bi);
            ce_out[r]  = ce;
            bin_out[r] = bi;
            atomicAdd(&shist[bi], 1u);
        }
        cur += 1; if (cur >= 3) cur -= 3;
    }

    __syncthreads();
    if (t < BINS) {
        unsigned c = shist[t];
        if (c) atomicAdd(&gcounts[t], c);       // integer atomics: deterministic
    }
}

// ---------------------------------------------------------------------------
// Kernel 0: zero histogram counters (ws is poisoned; must re-init every call)
// ---------------------------------------------------------------------------
__global__ void ghmc_init(unsigned* __restrict__ gcounts) {
    if (threadIdx.x < BINS) gcounts[threadIdx.x] = 0u;
}

// ---------------------------------------------------------------------------
// Kernel 2: weighted-sum partials. Each block recomputes the 10 bin weights
// and writes one partial into a fixed slot -> deterministic.
// ---------------------------------------------------------------------------
__global__ __launch_bounds__(BLK) void ghmc_pass2(
    const unsigned* __restrict__ gcounts, const float* __restrict__ ce,
    const int* __restrict__ bins, float* __restrict__ partial, int B)
{
    __shared__ float bw[BINS];
    __shared__ float red[8];
    const int t = threadIdx.x;
    if (t < BINS) {
        unsigned c = gcounts[t];
        bw[t] = c ? ((float)B) / (0.1f * (float)c) : 0.0f;  // tot/((1-mmt)*count)
    }
    __syncthreads();

    float s = 0.0f;
    for (int i = blockIdx.x * BLK + t; i < B; i += gridDim.x * BLK)
        s += bw[bins[i]] * ce[i];
    #pragma unroll
    for (int o = 16; o > 0; o >>= 1) s += __shfl_xor(s, o, 32);
    if ((t & 31) == 0) red[t >> 5] = s;
    __syncthreads();
    if (t == 0) {
        float tot = red[0];
        #pragma unroll
        for (int i = 1; i < 8; i++) tot += red[i];
        partial[blockIdx.x] = tot;
    }
}

// ---------------------------------------------------------------------------
// Kernel 3: fold partials serially (deterministic), divide by max(n,1)*B
// ---------------------------------------------------------------------------
__global__ void ghmc_final(const unsigned* __restrict__ gcounts,
                           const float* __restrict__ partial, int nPartial,
                           float* __restrict__ out, int B)
{
    if (threadIdx.x == 0 && blockIdx.x == 0) {
        int n = 0;
        for (int i = 0; i < BINS; i++) n += (gcounts[i] > 0u);
        float s = 0.0f;
        for (int i = 0; i < nPartial; i++) s += partial[i];
        float nn = (n < 1) ? 1.0f : (float)n;
        out[0] = s / (nn * (float)B);
    }
}

// ---------------------------------------------------------------------------
extern "C" void kernel_launch(void* const* d_in, const int* in_sizes, int n_in,
                              void* d_out, int out_size, void* d_ws, size_t ws_size,
                              hipStream_t stream) {
    (void)n_in; (void)out_size; (void)ws_size;
    const float* input  = (const float*)d_in[0];
    const float* target = (const float*)d_in[1];
    const int B = in_sizes[0] / C;              // 65536

    char* ws = (char*)d_ws;
    unsigned* counts  = (unsigned*)ws;                          // 40 B (BINS counters)
    float*    ce      = (float*)(ws + 128);                     // B floats
    int*      bins    = (int*)  (ws + 128 + (size_t)B * 4);     // B ints
    float*    partial = (float*)(ws + 128 + (size_t)B * 8);     // NB2 floats

    ghmc_init <<<1,   32,  0, stream>>>(counts);
    ghmc_pass1<<<NB1, BLK, 0, stream>>>(input, target, ce, bins, counts, B);
    ghmc_pass2<<<NB2, BLK, 0, stream>>>(counts, ce, bins, partial, B);
    ghmc_final<<<1,   32,  0, stream>>>(counts, partial, NB2, (float*)d_out, B);
}